// SimpleConvolution_17789754540876
// MI455X (gfx1250) — compile-verified
//
#include <hip/hip_runtime.h>

typedef __attribute__((ext_vector_type(2))) float v2f;
typedef __attribute__((ext_vector_type(4))) float v4f;
typedef __attribute__((ext_vector_type(8))) float v8f;

#define CIN_       128
#define COUT_      128
#define HW_        16384      // 128*128 pixels per (b, channel) plane
#define NBATCH_    32
#define PIX_PER_WG 64
// x tile stored K-pair-interleaved: lds[(row>>1)*XSTR + col*2 + (row&1)]
// XSTR = 128 data dwords + 32 pad -> pair p shifts banks by 32 (conflict-free
// half-wave split), and (row kb, row kb+1) of one column are one aligned b64.
#define XSTR       160

__global__ __launch_bounds__(256)
void conv1x1_wmma_f32(const float* __restrict__ x,
                      const float* __restrict__ w,
                      float* __restrict__ out) {
    __shared__ float lds[64 * XSTR];   // 64 K-pairs * 160 dwords = 40 KB

    const int    t  = threadIdx.x;
    const size_t P0 = (size_t)blockIdx.x * PIX_PER_WG;   // global pixel base
    const int    b  = (int)(P0 >> 14);                   // / HW_
    const int    p0 = (int)(P0 & (HW_ - 1));

    const float* xb   = x   + (size_t)b * CIN_  * HW_ + p0;
    float*       outb = out + (size_t)b * COUT_ * HW_ + p0;

    // ---- stage x tile (CIN_ x 64 px) into LDS, K-pair interleaved ----
    // 16 threads sweep a row (16B each -> 256B coalesced); each thread
    // interleaves rows (2p, 2p+1) and writes two b128 LDS stores.
    {
        const int q  = t & 15;         // column quad: cols 4q..4q+3
        const int pr = t >> 4;         // base pair 0..15
        #pragma unroll
        for (int s = 0; s < 4; ++s) {
            const int p   = pr + 16 * s;      // K-pair 0..63
            const int row = 2 * p;
            const int col = 4 * q;
            v4f r0 = __builtin_nontemporal_load(
                         (const v4f*)(xb + (size_t)row * HW_ + col));
            v4f r1 = __builtin_nontemporal_load(
                         (const v4f*)(xb + (size_t)(row + 1) * HW_ + col));
            v4f lo = {r0.x, r1.x, r0.y, r1.y};
            v4f hi4 = {r0.z, r1.z, r0.w, r1.w};
            *(v4f*)(&lds[p * XSTR + col * 2 + 0]) = lo;
            *(v4f*)(&lds[p * XSTR + col * 2 + 4]) = hi4;
        }
    }
    __syncthreads();

    const int lane  = t & 31;
    const int wave  = t >> 5;             // 0..7 -> COUT block
    const int obase = wave * 16;
    const int hi    = lane >> 4;          // lane half: ISA A/B layout +2 K rows
    const int lm    = lane & 15;

    // A fragment source (global, L2-hot): w[(kb + j + 2*hi)*COUT_ + obase + lm]
    const float* wbase = w + (size_t)(2 * hi) * COUT_ + obase + lm;
    // B fragment source: one b64 per (K-step, N-tile)
    //   v2f index = (kb/2 + hi)*80 + nt*16 + lm
    const v2f* bbase = (const v2f*)lds + (size_t)hi * (XSTR / 2) + lm;

    v8f acc[4];
    #pragma unroll
    for (int i = 0; i < 4; ++i) acc[i] = (v8f)0.0f;

    #pragma unroll
    for (int kc = 0; kc < 4; ++kc) {      // 4 chunks * 8 K-steps * K=4 = CIN_
        // prefetch A fragments for this chunk
        float a[16];
        #pragma unroll
        for (int u = 0; u < 8; ++u) {
            const int kb = (kc * 8 + u) * 4;
            a[u * 2 + 0] = wbase[(size_t)(kb + 0) * COUT_];
            a[u * 2 + 1] = wbase[(size_t)(kb + 1) * COUT_];
        }
        #pragma unroll
        for (int u = 0; u < 8; ++u) {
            const int kb = (kc * 8 + u) * 4;
            v2f av;
            av.x = a[u * 2 + 0];
            av.y = a[u * 2 + 1];
            #pragma unroll
            for (int nt = 0; nt < 4; ++nt) {
                const v2f bv = bbase[(kb >> 1) * (XSTR / 2) + nt * 16];
                acc[nt] = __builtin_amdgcn_wmma_f32_16x16x4_f32(
                    false, av, false, bv, (short)0, acc[nt], false, false);
            }
        }
    }

    // ---- store D tiles: VGPR r -> COUT rows obase + r + 8*hi, lanes -> pixels
    #pragma unroll
    for (int nt = 0; nt < 4; ++nt) {
        #pragma unroll
        for (int r = 0; r < 8; ++r) {
            const int orow = obase + r + 8 * hi;
            __builtin_nontemporal_store(
                acc[nt][r], outb + (size_t)orow * HW_ + nt * 16 + lm);
        }
    }
}

extern "C" void kernel_launch(void* const* d_in, const int* in_sizes, int n_in,
                              void* d_out, int out_size, void* d_ws, size_t ws_size,
                              hipStream_t stream) {
    const float* x = (const float*)d_in[0];   // (32, 128, 128, 128) fp32
    const float* w = (const float*)d_in[1];   // (128, 128) fp32
    float* out     = (float*)d_out;           // (32, 128, 128, 128) fp32

    const int total_pix = NBATCH_ * HW_;      // 524288
    dim3 grid(total_pix / PIX_PER_WG);        // 8192 workgroups
    conv1x1_wmma_f32<<<grid, 256, 0, stream>>>(x, w, out);
}